// LSGNN_56178172232393
// MI455X (gfx1250) — compile-verified
//
#include <hip/hip_runtime.h>
#include <hip/hip_bf16.h>
#include <math.h>

// ---------------------------------------------------------------------------
// LSGNN fused pipeline for MI455X (gfx1250, wave32, WMMA).
// Dominant cost: einsum('cnd,cdh->cnh') = 11x (1e5 x128)@(128x16) fp32 GEMMs,
// 563 MB streamed -> run on V_WMMA_F32_16X16X4_F32, keep all intermediates in
// LDS, finish with a second WMMA GEMM (96x40 padded to 96x48).
// ---------------------------------------------------------------------------

typedef float v2f __attribute__((ext_vector_type(2)));
typedef float v8f __attribute__((ext_vector_type(8)));

#define D_DIM   128
#define H_DIM   16
#define K_HOPS  5
#define C_DIM   11
#define OUT_DIM 40
#define KH      96   // (K+1)*H

// LDS padded strides (in floats). Even strides => 8-byte aligned b64 ds loads;
// chosen so consecutive lanes hit distinct banks.
#define WT_STRIDE    132   // wT[n][k],   n<16,  k<128   (132 mod 64 = 4)
#define WOT_STRIDE   98    // woT[col][k], col<48, k<96  (98 mod 64 = 34)
#define OB_M_STRIDE  18    // outbuf[c][m][n], m stride
#define OB_C_STRIDE  (16*OB_M_STRIDE)  // 288
#define FU_STRIDE    98    // fused[m][k], k<96

__device__ __forceinline__ float silu_f(float x) { return x / (1.0f + __expf(-x)); }

// ---------------------------------------------------------------------------
__global__ void zero_kernel(float* __restrict__ p, int n) {
  int i = blockIdx.x * blockDim.x + threadIdx.x;
  if (i < n) p[i] = 0.0f;
}

// ---------------------------------------------------------------------------
// Per-edge MLP 2->16->1 (weights are uniform -> scalar-cached), scatter-add.
__global__ void edge_kernel(const long long* __restrict__ ei,   // (2,E) int64
                            const float* __restrict__ dist,     // (E,2)
                            const float* __restrict__ Wd1,      // (2,16)
                            const float* __restrict__ bd1,      // (16)
                            const float* __restrict__ Wd2,      // (16,1)
                            const float* __restrict__ bd2,      // (1)
                            float* __restrict__ seg, float* __restrict__ cnt,
                            int E) {
  int e = blockIdx.x * blockDim.x + threadIdx.x;
  if (e >= E) return;
  float d0 = dist[2 * e + 0];
  float d1 = dist[2 * e + 1];
  float acc = bd2[0];
#pragma unroll
  for (int j = 0; j < 16; ++j) {
    float h = silu_f(d0 * Wd1[j] + d1 * Wd1[16 + j] + bd1[j]);
    acc += h * Wd2[j];
  }
  long long t = ei[(long long)E + e];  // edge_index[1][e]
  atomicAdd(&seg[t], acc);
  atomicAdd(&cnt[t], 1.0f);
}

// ---------------------------------------------------------------------------
// Per-node alpha MLP: [ls, ls^2] -> 16 (silu) -> 15.
__global__ void alpha_kernel(const float* __restrict__ seg,
                             const float* __restrict__ cnt,
                             const float* __restrict__ Wa1,  // (2,16)
                             const float* __restrict__ ba1,  // (16)
                             const float* __restrict__ Wa2,  // (16,15)
                             const float* __restrict__ ba2,  // (15)
                             float* __restrict__ alpha,      // (N,15)
                             int N) {
  int i = blockIdx.x * blockDim.x + threadIdx.x;
  if (i >= N) return;
  float c = cnt[i];
  float ls = seg[i] / fmaxf(c, 1.0f);
  float t0 = ls, t1 = ls * ls;
  float h[16];
#pragma unroll
  for (int j = 0; j < 16; ++j)
    h[j] = silu_f(t0 * Wa1[j] + t1 * Wa1[16 + j] + ba1[j]);
#pragma unroll
  for (int o = 0; o < 15; ++o) {
    float a = ba2[o];
#pragma unroll
    for (int j = 0; j < 16; ++j) a += h[j] * Wa2[j * 15 + o];
    alpha[(long long)i * 15 + o] = a;
  }
}

// ---------------------------------------------------------------------------
// l2-normalize an 8-float half-vector; partner lane (lane^1) holds the other
// 8 channels of the same node -> pair-reduce via shfl_xor.
__device__ __forceinline__ void l2norm8(float v[8]) {
  float ss = 0.0f;
#pragma unroll
  for (int j = 0; j < 8; ++j) ss += v[j] * v[j];
  ss += __shfl_xor(ss, 1, 32);
  float inv = 1.0f / fmaxf(sqrtf(ss), 1e-12f);
#pragma unroll
  for (int j = 0; j < 8; ++j) v[j] *= inv;
}

// ---------------------------------------------------------------------------
__launch_bounds__(64)
__global__ void main_kernel(const float* __restrict__ xolh,   // (C,N,128)
                            const float* __restrict__ rw,     // (C,128,16)
                            const float* __restrict__ Wo,     // (96,40)
                            const float* __restrict__ bo,     // (40)
                            const float* __restrict__ alpha,  // (N,15)
                            float* __restrict__ out,          // (N,40)
                            int N) {
  __shared__ __align__(16) float wT[16 * WT_STRIDE];          //  8448 B
  __shared__ __align__(16) float woT[48 * WOT_STRIDE];        // 18816 B
  __shared__ __align__(16) float outbuf[2][C_DIM * OB_C_STRIDE]; // 25344 B
  __shared__ __align__(16) float fusedb[2][16 * FU_STRIDE];   // 12544 B

  const int tid  = threadIdx.x;
  const int wave = tid >> 5;
  const int lane = tid & 31;
  const int mrow = lane & 15;           // WMMA: A row M / B,D column N
  const int h2   = (lane >> 4) << 1;    // K sub-offset: 0 (lanes 0-15) / 2
  const int mhi  = (lane >> 4) * 8;     // D-matrix M offset for this half

  int node_base = (blockIdx.x * 2 + wave) * 16;
  if (node_base + 16 > N) node_base = N - 16;  // N%16==0 assumed (N=100000)

  // Stage zero-padded Wo^T once per block: woT[col][k] = Wo[k][col], col<48.
  for (int idx = tid; idx < 48 * KH; idx += 64) {
    int k = idx / 48, col = idx % 48;
    woT[col * WOT_STRIDE + k] = (col < OUT_DIM) ? Wo[k * OUT_DIM + col] : 0.0f;
  }

  float* myout = outbuf[wave];

  // ---- main einsum: per c, (16x128)@(128x16) via 32x WMMA f32 16x16x4 ----
  for (int c = 0; c < C_DIM; ++c) {
    __syncthreads();  // previous c's wT fully consumed
    for (int idx = tid; idx < D_DIM * H_DIM; idx += 64) {
      int k = idx >> 4, n = idx & 15;           // coalesced global read
      wT[n * WT_STRIDE + k] = rw[(c * D_DIM + k) * H_DIM + n];
    }
    __syncthreads();

    const float* Abase = xolh + ((size_t)c * N + node_base + mrow) * D_DIM + h2;
    const float* Bbase = wT + mrow * WT_STRIDE + h2;
    v8f acc = {};
#pragma unroll 8
    for (int k0 = 0; k0 < D_DIM; k0 += 4) {
      v2f a = __builtin_nontemporal_load((const v2f*)(Abase + k0)); // stream
      v2f b = *(const v2f*)(Bbase + k0);
      acc = __builtin_amdgcn_wmma_f32_16x16x4_f32(
          false, a, false, b, (short)0, acc, false, false);
    }
    // D layout: reg j holds [M = j + 8*(lane/16)][Ncol = lane%16]
#pragma unroll
    for (int j = 0; j < 8; ++j)
      myout[c * OB_C_STRIDE + (j + mhi) * OB_M_STRIDE + mrow] = acc[j];
  }

  // ---- per-node normalize + fuse (2 lanes per node, 8 channels each) ----
  const int   m  = lane >> 1;
  const int   co = (lane & 1) * 8;
  const float* ob = myout + m * OB_M_STRIDE + co;
  float* fb = fusedb[wave];

  float x0n[8];
#pragma unroll
  for (int j = 0; j < 8; ++j) x0n[j] = ob[0 * OB_C_STRIDE + j];
  l2norm8(x0n);
#pragma unroll
  for (int j = 0; j < 8; ++j) x0n[j] = fmaxf(x0n[j], 0.0f);  // relu(l2norm)

  {  // fused row 0 = l2norm(x0n)
    float r0[8];
#pragma unroll
    for (int j = 0; j < 8; ++j) r0[j] = x0n[j];
    l2norm8(r0);
#pragma unroll
    for (int j = 0; j < 8; ++j) fb[m * FU_STRIDE + co + j] = r0[j];
  }

  const float* ap = alpha + (size_t)(node_base + m) * 15;
#pragma unroll
  for (int k = 1; k <= K_HOPS; ++k) {
    float L8[8], H8[8];
#pragma unroll
    for (int j = 0; j < 8; ++j) L8[j] = ob[k * OB_C_STRIDE + j];
    l2norm8(L8);
#pragma unroll
    for (int j = 0; j < 8; ++j) L8[j] = fmaxf(L8[j], 0.0f);
#pragma unroll
    for (int j = 0; j < 8; ++j) H8[j] = ob[(K_HOPS + k) * OB_C_STRIDE + j];
    l2norm8(H8);
#pragma unroll
    for (int j = 0; j < 8; ++j) H8[j] = fmaxf(H8[j], 0.0f);

    float aI = ap[(k - 1) * 3 + 0];
    float aL = ap[(k - 1) * 3 + 1];
    float aH = ap[(k - 1) * 3 + 2];
    float f[8];
#pragma unroll
    for (int j = 0; j < 8; ++j) f[j] = aI * x0n[j] + aL * L8[j] + aH * H8[j];
    l2norm8(f);
#pragma unroll
    for (int j = 0; j < 8; ++j) fb[m * FU_STRIDE + k * H_DIM + co + j] = f[j];
  }

  // ---- final GEMM: fused(16x96) @ Wo(96x48 padded), 3 n-tiles x 24 WMMA ----
  // fb is wave-private; same-wave LDS ops are in-order, no block barrier needed.
  const float* Af = fb + mrow * FU_STRIDE + h2;
#pragma unroll
  for (int nt = 0; nt < 3; ++nt) {
    const float* Bf = woT + (nt * 16 + mrow) * WOT_STRIDE + h2;
    v8f acc = {};
#pragma unroll
    for (int k0 = 0; k0 < KH; k0 += 4) {
      v2f a = *(const v2f*)(Af + k0);
      v2f b = *(const v2f*)(Bf + k0);
      acc = __builtin_amdgcn_wmma_f32_16x16x4_f32(
          false, a, false, b, (short)0, acc, false, false);
    }
    int n = nt * 16 + mrow;
    if (n < OUT_DIM) {           // divergence only after WMMA (EXEC all-1 rule)
      float bias = bo[n];
#pragma unroll
      for (int j = 0; j < 8; ++j)
        out[(size_t)(node_base + j + mhi) * OUT_DIM + n] = acc[j] + bias;
    }
  }
}

// ---------------------------------------------------------------------------
extern "C" void kernel_launch(void* const* d_in, const int* in_sizes, int n_in,
                              void* d_out, int out_size, void* d_ws, size_t ws_size,
                              hipStream_t stream) {
  const float*     x    = (const float*)d_in[0];   (void)x;  // shape-only
  const long long* ei   = (const long long*)d_in[1];
  const float*     dist = (const float*)d_in[2];
  const float*     xolh = (const float*)d_in[3];
  const float*     Wd1  = (const float*)d_in[4];
  const float*     bd1  = (const float*)d_in[5];
  const float*     Wd2  = (const float*)d_in[6];
  const float*     bd2  = (const float*)d_in[7];
  const float*     Wa1  = (const float*)d_in[8];
  const float*     ba1  = (const float*)d_in[9];
  const float*     Wa2  = (const float*)d_in[10];
  const float*     ba2  = (const float*)d_in[11];
  const float*     rw   = (const float*)d_in[12];
  const float*     Wo   = (const float*)d_in[13];
  const float*     bo   = (const float*)d_in[14];
  float* out = (float*)d_out;
  (void)n_in; (void)out_size; (void)ws_size;

  const int N = in_sizes[0] / D_DIM;   // 100000
  const int E = in_sizes[2] / 2;       // 3200000

  // Workspace: seg_sum[N] | seg_cnt[N] | alpha[N*15]  (17N floats ~ 6.8 MB)
  float* seg   = (float*)d_ws;
  float* cnt   = seg + N;
  float* alpha = cnt + N;

  zero_kernel<<<(2 * N + 255) / 256, 256, 0, stream>>>(seg, 2 * N);
  edge_kernel<<<(E + 255) / 256, 256, 0, stream>>>(ei, dist, Wd1, bd1, Wd2, bd2,
                                                   seg, cnt, E);
  alpha_kernel<<<(N + 255) / 256, 256, 0, stream>>>(seg, cnt, Wa1, ba1, Wa2,
                                                    ba2, alpha, N);

  const int blocks = (N + 31) / 32;  // 2 waves/block, 16 nodes/wave
  main_kernel<<<blocks, 64, 0, stream>>>(xolh, rw, Wo, bo, alpha, out, N);
}